// FFM_73169062855073
// MI455X (gfx1250) — compile-verified
//
#include <hip/hip_runtime.h>
#include <math.h>

// FFM forward: out[b] = sigmoid( sum_f w1[idxs[b,f]]*vals[b,f]
//                              + sum_{i<j<=37} dot(emb[i,idxs[b,j]], emb[j,idxs[b,i]]) * vals[b,i]*vals[b,j] )
//
// Gather-bound: ~115 MB of 40-B row gathers, table (156 MB) resident in 192 MB L2.
// One wave32 per batch element; WMMA f32 16x16x4 used as an exact, zero-cost wave reduction.

typedef __attribute__((ext_vector_type(2))) float v2f;
typedef __attribute__((ext_vector_type(8))) float v8f;

#define NFIELD 39
#define NPAIRS 703   // C(38,2): pairs (i,j), i<j<=37
#define EMB    10
#define FEAT   100000
#define WAVES_PER_BLOCK 8

__global__ __launch_bounds__(256) void ffm_fwd_kernel(
    const int*   __restrict__ idxs,   // [B,39]
    const float* __restrict__ vals,   // [B,39]
    const float* __restrict__ emb,    // [39,100000,10]
    const float* __restrict__ w1,     // [100000]
    float*       __restrict__ out,    // [B]
    int B)
{
    __shared__ unsigned short s_pair[NPAIRS];       // (i<<8)|j
    __shared__ int   s_idx[WAVES_PER_BLOCK][40];    // per-wave staged row indices
    __shared__ float s_val[WAVES_PER_BLOCK][40];    // per-wave staged row values

    const int tid  = threadIdx.x;
    const int lane = tid & 31;
    const int w    = tid >> 5;
    const int b    = blockIdx.x * WAVES_PER_BLOCK + w;   // one wave per sample

    // ---- block-shared pair LUT: decode linear pair id -> (i,j) ----
    for (int t = tid; t < NPAIRS; t += blockDim.x) {
        int i = 0, rem = t;
        // row i contributes (37 - i) pairs: j in [i+1, 37]
        while (rem >= (37 - i)) { rem -= (37 - i); ++i; }
        int j = i + 1 + rem;
        s_pair[t] = (unsigned short)((i << 8) | j);
    }

    // ---- stage this sample's indices & values into LDS ----
    if (b < B) {
        const long base = (long)b * NFIELD;
        int f = lane;
        if (f < NFIELD) { s_idx[w][f] = idxs[base + f]; s_val[w][f] = vals[base + f]; }
        f = lane + 32;
        if (f < NFIELD) { s_idx[w][f] = idxs[base + f]; s_val[w][f] = vals[base + f]; }
    }
    __syncthreads();

    float acc = 0.0f;

    if (b < B) {
        // ---- second-order pairwise gathers + dot products ----
        for (int p = lane; p < NPAIRS; p += 32) {
            const int pr = s_pair[p];
            const int i  = pr >> 8;
            const int j  = pr & 0xff;
            const int row_ij = s_idx[w][j];     // table i gathered with idxs[:, j]
            const int row_ji = s_idx[w][i];     // table j gathered with idxs[:, i]
            const float vi = s_val[w][i];
            const float vj = s_val[w][j];
            // rows are 40-byte multiples from the allocation base -> 8-byte aligned
            const v2f* __restrict__ pa =
                (const v2f*)(emb + ((size_t)i * (size_t)FEAT + (size_t)row_ij) * (size_t)EMB);
            const v2f* __restrict__ pb =
                (const v2f*)(emb + ((size_t)j * (size_t)FEAT + (size_t)row_ji) * (size_t)EMB);
            float dot = 0.0f;
            #pragma unroll
            for (int k = 0; k < EMB / 2; ++k) {
                v2f av = pa[k];
                v2f bv = pb[k];
                dot += av.x * bv.x;
                dot += av.y * bv.y;
            }
            acc += dot * vi * vj;
        }

        // ---- first-order term folded into the same accumulator ----
        int f = lane;
        if (f < NFIELD) acc += w1[s_idx[w][f]] * s_val[w][f];
        f = lane + 32;
        if (f < NFIELD) acc += w1[s_idx[w][f]] * s_val[w][f];
    }

    // ---- exact f32 wave reduction via V_WMMA_F32_16X16X4_F32 ----
    // A[lane] = {acc, 0}; B = all ones  =>  D[m][n] = rowsum_m(A).
    // Sum of a lane's 8 D regs covers M-half {0..7} (lanes 0-15) or {8..15} (lanes 16-31);
    // adding the lane^16 partner's half-sum yields sum over all 64 A entries = wave total.
    // EXEC is all-1s here (wave-uniform control flow) as the ISA requires.
    v2f amat; amat.x = acc;  amat.y = 0.0f;
    v2f ones; ones.x = 1.0f; ones.y = 1.0f;
    v8f cmat = {0.0f, 0.0f, 0.0f, 0.0f, 0.0f, 0.0f, 0.0f, 0.0f};
    v8f d = __builtin_amdgcn_wmma_f32_16x16x4_f32(
        /*neg_a=*/false, amat, /*neg_b=*/false, ones,
        /*c_mod=*/(short)0, cmat, /*reuse_a=*/false, /*reuse_b=*/false);
    float half_sum = d[0] + d[1] + d[2] + d[3] + d[4] + d[5] + d[6] + d[7];
    float total = half_sum + __shfl_xor(half_sum, 16, 32);

    if (b < B && lane == 0) {
        out[b] = 1.0f / (1.0f + expf(-total));
    }
}

extern "C" void kernel_launch(void* const* d_in, const int* in_sizes, int n_in,
                              void* d_out, int out_size, void* d_ws, size_t ws_size,
                              hipStream_t stream) {
    const int*   idxs = (const int*)  d_in[0];   // [B,39] int32
    const float* vals = (const float*)d_in[1];   // [B,39] f32
    const float* emb  = (const float*)d_in[2];   // [39,100000,10] f32
    const float* w1   = (const float*)d_in[3];   // [100000,1] f32
    float*       out  = (float*)d_out;           // [B] f32

    const int B = in_sizes[0] / NFIELD;
    const int blocks = (B + WAVES_PER_BLOCK - 1) / WAVES_PER_BLOCK;
    ffm_fwd_kernel<<<blocks, 256, 0, stream>>>(idxs, vals, emb, w1, out, B);
}